// CenterLossLayer_2654289789633
// MI455X (gfx1250) — compile-verified
//
#include <hip/hip_runtime.h>

#define ALPHA       0.5f
#define NUM_CLASSES 100
#define DIM         512
#define BATCH       65536

typedef float v2f __attribute__((ext_vector_type(2)));
typedef float v4f __attribute__((ext_vector_type(4)));
typedef float v8f __attribute__((ext_vector_type(8)));

// ---------------------------------------------------------------------------
// ws layout (floats): [0, 51200) per-class feature sums, [51200, 51300) counts
// ---------------------------------------------------------------------------

__global__ void k_zero(float* __restrict__ ws, int n) {
    int i = blockIdx.x * blockDim.x + threadIdx.x;
    if (i < n) ws[i] = 0.0f;
}

// ---------------------------------------------------------------------------
// Per-class counts = onehot^T @ ones, via V_WMMA_F32_16X16X4_F32.
// A (16x4 f32, documented layout): lane 0-15 -> M=lane, lane 16-31 -> M=lane-16,
//   VGPR v holds K = 2*(lane>>4) + v.  B = all-ones (layout independent).
// D (16x16 f32, documented): VGPR r, lanes 0-15 -> M=r,N=lane; lanes 16-31 -> M=r+8.
// Classes >= 100 alias the class-0 column (in bounds); their D rows are simply
// never extracted, so no per-load masking (and no EXEC divergence) is needed.
// ---------------------------------------------------------------------------
#define CG      7      // class groups of 16 (covers 112 >= 100)
#define SLICES  32     // batch slices
#define CWAVES  8      // waves per block
#define RPW     (BATCH / SLICES / CWAVES)   // rows per wave = 256

__global__ void k_count(const float* __restrict__ onehot, float* __restrict__ gcnt) {
    const int cg    = blockIdx.x % CG;
    const int slice = blockIdx.x / CG;
    const int wave  = threadIdx.x >> 5;
    const int lane  = threadIdx.x & 31;
    const int c0    = cg * 16;
    const int c     = c0 + (lane & 15);
    const int csafe = (c < NUM_CLASSES) ? c : 0;

    const int row0 = slice * (BATCH / SLICES) + wave * RPW + ((lane >> 4) << 1);

    const float* p = onehot + (size_t)row0 * NUM_CLASSES + csafe;

    v8f d = {};
    v2f bones = {1.0f, 1.0f};

    for (int k = 0; k < RPW; k += 4) {
        v2f a;
        a.x = p[0];
        a.y = p[NUM_CLASSES];
        d = __builtin_amdgcn_wmma_f32_16x16x4_f32(false, a, false, bones,
                                                  (short)0, d, false, false);
        p += 4 * NUM_CLASSES;
    }

    if ((lane & 15) == 0) {           // N == 0 column: lanes 0 and 16
        const int mbase = (lane >> 4) * 8;
#pragma unroll
        for (int r = 0; r < 8; ++r) {
            int cc = c0 + mbase + r;
            if (cc < NUM_CLASSES) unsafeAtomicAdd(&gcnt[cc], d[r]);
        }
    }
}

// ---------------------------------------------------------------------------
// Main streaming pass: one wave owns one batch row at a time.
//  - label via ballot over the onehot row (clamped loads, masked predicate)
//  - squared distance to centers[label]: non-temporal float4 loads + shfl reduce
//  - scatter features into a 200KB per-class LDS accumulator (ds_add_f32;
//    the whole 100x512 f32 accumulator fits in CDNA5's 320KB WGP LDS)
//  - block-level combine into global partial sums
// ---------------------------------------------------------------------------
#define MB 128   // blocks (1 per WGP; 200KB dynamic LDS each)
#define MT 512   // threads per block = 16 waves

__global__ void k_main(const float* __restrict__ feat,
                       const float* __restrict__ onehot,
                       const float* __restrict__ centers,
                       float* __restrict__ result,
                       float* __restrict__ gsum) {
    extern __shared__ float acc[];    // NUM_CLASSES * DIM floats
    const int tid = threadIdx.x;

    for (int i = tid; i < NUM_CLASSES * DIM; i += MT) acc[i] = 0.0f;
    __syncthreads();

    const int lane   = tid & 31;
    const int gwave  = blockIdx.x * (MT / 32) + (tid >> 5);
    const int nwaves = MB * (MT / 32);

    for (int row = gwave; row < BATCH; row += nwaves) {
        // prefetch this wave's next feature row (global_prefetch_b8)
        int nrow = row + nwaves;
        if (nrow < BATCH)
            __builtin_prefetch(feat + (size_t)nrow * DIM + lane * 16, 0, 1);

        // --- find label: scan the 100-wide onehot row in 4 chunks of 32 ---
        const float* oh = onehot + (size_t)row * NUM_CLASSES;
        int lab = 0;
#pragma unroll
        for (int chunk = 0; chunk < 4; ++chunk) {
            int idx  = chunk * 32 + lane;
            int safe = (idx < NUM_CLASSES) ? idx : (NUM_CLASSES - 1);
            float v  = oh[safe];
            unsigned long long m = __ballot((v > 0.5f) & (idx < NUM_CLASSES));
            if (m != 0ull) lab = chunk * 32 + (__ffsll(m) - 1);
        }

        const float* f = feat    + (size_t)row * DIM;
        const float* c = centers + (size_t)lab * DIM;
        float*       a = acc     + lab * DIM;

        float sq = 0.0f;
#pragma unroll
        for (int j = 0; j < 4; ++j) {
            int d0 = j * 128 + lane * 4;
            v4f fv = __builtin_nontemporal_load((const v4f*)(f + d0));
            v4f cv = *(const v4f*)(c + d0);
            v4f dv = fv - cv;
            sq += dv.x * dv.x + dv.y * dv.y + dv.z * dv.z + dv.w * dv.w;
            unsafeAtomicAdd(&a[d0 + 0], fv.x);
            unsafeAtomicAdd(&a[d0 + 1], fv.y);
            unsafeAtomicAdd(&a[d0 + 2], fv.z);
            unsafeAtomicAdd(&a[d0 + 3], fv.w);
        }
#pragma unroll
        for (int off = 16; off > 0; off >>= 1) sq += __shfl_down(sq, off, 32);
        if (lane == 0) __builtin_nontemporal_store(sq, &result[row]);
    }

    __syncthreads();
    // combine this block's partial class sums into the global accumulator
    for (int i = tid; i < NUM_CLASSES * DIM; i += MT) {
        float v = acc[i];
        if (v != 0.0f) unsafeAtomicAdd(&gsum[i], v);
    }
}

// ---------------------------------------------------------------------------
// new_centers[c,d] = centers - ALPHA * (cnt*centers - featsum) / (cnt + 1)
// ---------------------------------------------------------------------------
__global__ void k_final(const float* __restrict__ centers,
                        const float* __restrict__ gsum,
                        const float* __restrict__ gcnt,
                        float* __restrict__ newc) {
    int i = blockIdx.x * blockDim.x + threadIdx.x;
    if (i >= NUM_CLASSES * DIM) return;
    int cls   = i / DIM;
    float cnt = gcnt[cls];
    float ctr = centers[i];
    float delta = (cnt * ctr - gsum[i]) / (cnt + 1.0f);
    newc[i] = ctr - ALPHA * delta;
}

// ---------------------------------------------------------------------------

extern "C" void kernel_launch(void* const* d_in, const int* in_sizes, int n_in,
                              void* d_out, int out_size, void* d_ws, size_t ws_size,
                              hipStream_t stream) {
    (void)in_sizes; (void)n_in; (void)out_size; (void)ws_size;

    const float* feat    = (const float*)d_in[0];   // [BATCH, DIM]
    const float* onehot  = (const float*)d_in[1];   // [BATCH, NUM_CLASSES]
    const float* centers = (const float*)d_in[2];   // [NUM_CLASSES, DIM]

    float* result = (float*)d_out;                  // [BATCH]
    float* newc   = (float*)d_out + BATCH;          // [NUM_CLASSES, DIM]

    float* gsum = (float*)d_ws;                     // [NUM_CLASSES*DIM]
    float* gcnt = gsum + NUM_CLASSES * DIM;         // [NUM_CLASSES]

    const int nz = NUM_CLASSES * DIM + NUM_CLASSES;
    k_zero<<<(nz + 255) / 256, 256, 0, stream>>>((float*)d_ws, nz);

    k_count<<<CG * SLICES, 32 * CWAVES, 0, stream>>>(onehot, gcnt);

    const int lds_bytes = NUM_CLASSES * DIM * (int)sizeof(float);  // 204800
    (void)hipFuncSetAttribute((const void*)k_main,
                              hipFuncAttributeMaxDynamicSharedMemorySize, lds_bytes);
    k_main<<<MB, MT, lds_bytes, stream>>>(feat, onehot, centers, result, gsum);

    k_final<<<(NUM_CLASSES * DIM + 255) / 256, 256, 0, stream>>>(centers, gsum, gcnt, newc);
}